// RegressionModel_65481071407369
// MI455X (gfx1250) — compile-verified
//
#include <hip/hip_runtime.h>

typedef __attribute__((ext_vector_type(2))) float v2f;
typedef __attribute__((ext_vector_type(8))) float v8f;

#define QDIM     16384   // 2^14 amplitudes
#define NWIRES   14
#define NLAYERS  2
#define THREADS  512     // 16 wave32s
#define NWAVES   (THREADS / 32)
#define NTILES   (QDIM / 256)          // 64 16x16 prob tiles
#define TILES_PER_WAVE (NTILES / NWAVES)

// One workgroup = one batch element. Full complex statevector lives in LDS
// (16384 * 8B = 128 KB of the WGP's 320 KB -> 2 blocks per WGP), so the whole
// 2-layer circuit runs at LDS bandwidth; HBM is touched once in, once out.
// Each (wire) step is ONE fused quad pass: RZ*RY on wire w, CNOT(w,w+1), RX(w).
extern "C" __global__ void __launch_bounds__(THREADS)
qsim14_kernel(const float* __restrict__ state,    // (256, 16384) f32
              const float* __restrict__ params,   // (2, 14, 3)  f32
              const float* __restrict__ head_w,   // (1, 14)     f32
              const float* __restrict__ head_b,   // (1,)        f32
              float* __restrict__ out)            // (256,)      f32
{
    extern __shared__ char smem_raw[];
    float2* psi   = (float2*)smem_raw;                          // 131072 B
    float*  feats = (float*)(smem_raw + QDIM * sizeof(float2)); // 16 floats

    const int tid = threadIdx.x;
    const int b   = blockIdx.x;

    // ---- load real input state, imag = 0 (coalesced float4 reads) ----
    {
        const float4* src4 = (const float4*)(state + (size_t)b * QDIM);
        #pragma unroll
        for (int j = 0; j < QDIM / (THREADS * 4); ++j) {   // 8 iters
            int i4 = tid + j * THREADS;
            float4 v = src4[i4];
            int ib = i4 * 4;
            psi[ib + 0] = make_float2(v.x, 0.0f);
            psi[ib + 1] = make_float2(v.y, 0.0f);
            psi[ib + 2] = make_float2(v.z, 0.0f);
            psi[ib + 3] = make_float2(v.w, 0.0f);
        }
        if (tid < 16) feats[tid] = 0.0f;
    }
    __syncthreads();

    // ---- circuit: one fused quad pass per (layer, wire) ----
    for (int l = 0; l < NLAYERS; ++l) {
        for (int w = 0; w < NWIRES; ++w) {
            const float t0 = params[(l * NWIRES + w) * 3 + 0];
            const float t1 = params[(l * NWIRES + w) * 3 + 1];
            const float t2 = params[(l * NWIRES + w) * 3 + 2];
            const float cy = cosf(0.5f * t0), sy = sinf(0.5f * t0);
            const float cz = cosf(0.5f * t1), sz = sinf(0.5f * t1);
            const float cx = cosf(0.5f * t2), sx = sinf(0.5f * t2);

            const int w2 = (w + 1) % NWIRES;
            const int pc = 13 - w, pt = 13 - w2;   // wire 0 is MSB axis
            const int hi = pc > pt ? pc : pt;
            const int lo = pc > pt ? pt : pc;
            const int mlo = (1 << lo) - 1, mhi = (1 << hi) - 1;
            const int sc = 1 << pc, st = 1 << pt;

            #pragma unroll 4
            for (int r = 0; r < QDIM / 4 / THREADS; ++r) {   // 8 quads/thread
                const int q = tid + r * THREADS;
                const int u = ((q & ~mlo) << 1) | (q & mlo);
                const int i = ((u & ~mhi) << 1) | (u & mhi);
                const float2 a00 = psi[i];
                const float2 a01 = psi[i + st];
                const float2 a10 = psi[i + sc];
                const float2 a11 = psi[i + sc + st];

                // RZ(t1)*RY(t0) on wire w (mixes first-bit pairs):
                //   u0 = e^{-i t1/2} (cy*a0 - sy*a1) ; u1 = e^{+i t1/2} (sy*a0 + cy*a1)
                const float r0r = cy * a00.x - sy * a10.x, r0i = cy * a00.y - sy * a10.y;
                const float r1r = sy * a00.x + cy * a10.x, r1i = sy * a00.y + cy * a10.y;
                const float r2r = cy * a01.x - sy * a11.x, r2i = cy * a01.y - sy * a11.y;
                const float r3r = sy * a01.x + cy * a11.x, r3i = sy * a01.y + cy * a11.y;
                const float2 u00 = make_float2(cz * r0r + sz * r0i, cz * r0i - sz * r0r);
                const float2 u10 = make_float2(cz * r1r - sz * r1i, cz * r1i + sz * r1r);
                const float2 u01 = make_float2(cz * r2r + sz * r2i, cz * r2i - sz * r2r);
                const float2 u11 = make_float2(cz * r3r - sz * r3i, cz * r3i + sz * r3r);

                // CNOT (swap u10<->u11) then RX(t2) on wire w; mis(z)=(sx*z.y,-sx*z.x)
                psi[i]           = make_float2(cx * u00.x + sx * u11.y,
                                               cx * u00.y - sx * u11.x);
                psi[i + sc]      = make_float2(sx * u00.y + cx * u11.x,
                                              -sx * u00.x + cx * u11.y);
                psi[i + st]      = make_float2(cx * u01.x + sx * u10.y,
                                               cx * u01.y - sx * u10.x);
                psi[i + sc + st] = make_float2(sx * u01.y + cx * u10.x,
                                              -sx * u01.x + cx * u10.y);
            }
            __syncthreads();
        }
    }

    // ---- feature extraction: feats[w] = sum_i |psi_i|^2 * (1 - 2*bit_w(i)) ----

    // wires 0..5: sign constant over each thread's 32-aligned contiguous chunk
    {
        const int base = tid * (QDIM / THREADS);   // 32 amplitudes per thread
        float csum = 0.0f;
        #pragma unroll 8
        for (int r = 0; r < QDIM / THREADS; ++r) {
            const float2 z = psi[base + r];
            csum += z.x * z.x + z.y * z.y;
        }
        #pragma unroll
        for (int wq = 0; wq < 6; ++wq) {
            const float sgn = ((base >> (13 - wq)) & 1) ? -1.0f : 1.0f;
            atomicAdd(&feats[wq], sgn * csum);
        }
    }

    // wires 6..13 via WMMA: per 16x16 tile (k bits -> wires 10..13,
    // n bits -> wires 6..9): D = A(signs | ones | 0) x P, K=16 as 4 chained
    // V_WMMA_F32_16X16X4_F32; accumulator carried over the wave's tiles.
    {
        const int lane = tid & 31;
        const int wave = tid >> 5;
        const int half = lane >> 4;       // 0: lanes 0-15, 1: lanes 16-31
        const int mrow = lane & 15;       // A: M = lane%16
        const int ncol = lane & 15;       // B: N = lane%16

        // A element layout per ISA: VGPR0 holds K = base+2*half, VGPR1 K+1
        float a0v[4], a1v[4];
        #pragma unroll
        for (int kk = 0; kk < 4; ++kk) {
            const int k0 = kk * 4 + 2 * half;
            const int k1 = k0 + 1;
            a0v[kk] = (mrow < 4) ? (((k0 >> mrow) & 1) ? -1.0f : 1.0f)
                                 : (mrow == 4 ? 1.0f : 0.0f);
            a1v[kk] = (mrow < 4) ? (((k1 >> mrow) & 1) ? -1.0f : 1.0f)
                                 : (mrow == 4 ? 1.0f : 0.0f);
        }

        v8f acc = {};
        const float4* psi4 = (const float4*)psi;   // two float2 per float4
        #pragma unroll
        for (int t = 0; t < TILES_PER_WAVE; ++t) { // uniform trip count
            const int o = wave + t * NWAVES;
            const int ibase = o * 256 + ncol * 16;
            #pragma unroll
            for (int kk = 0; kk < 4; ++kk) {
                const int k0 = kk * 4 + 2 * half;          // even
                const float4 qv = psi4[(ibase + k0) >> 1]; // |psi|^2 pair
                v2f A = { a0v[kk], a1v[kk] };
                v2f B = { qv.x * qv.x + qv.y * qv.y,
                          qv.z * qv.z + qv.w * qv.w };
                acc = __builtin_amdgcn_wmma_f32_16x16x4_f32(
                        /*neg_a=*/false, A, /*neg_b=*/false, B,
                        /*c_mod=*/(short)0, acc,
                        /*reuse_a=*/false, /*reuse_b=*/false);
            }
        }

        if (lane < 16) {   // rows 0..4 live in acc[0..4] on lanes 0..15
            #pragma unroll
            for (int m = 0; m < 4; ++m)
                atomicAdd(&feats[13 - m], acc[m]);         // wires 13..10
            const float cs = acc[4];                        // column sums
            #pragma unroll
            for (int j = 0; j < 4; ++j) {                   // wires 9..6
                const float sgn = ((lane >> j) & 1) ? -1.0f : 1.0f;
                atomicAdd(&feats[9 - j], sgn * cs);
            }
        }
    }
    __syncthreads();

    // ---- linear head ----
    if (tid == 0) {
        float r = head_b[0];
        #pragma unroll
        for (int w = 0; w < NWIRES; ++w) r += feats[w] * head_w[w];
        out[b] = r;
    }
}

extern "C" void kernel_launch(void* const* d_in, const int* in_sizes, int n_in,
                              void* d_out, int out_size, void* d_ws, size_t ws_size,
                              hipStream_t stream) {
    (void)in_sizes; (void)n_in; (void)d_ws; (void)ws_size; (void)out_size;
    const float* state  = (const float*)d_in[0];
    const float* params = (const float*)d_in[1];
    const float* head_w = (const float*)d_in[2];
    const float* head_b = (const float*)d_in[3];
    float* out = (float*)d_out;

    const size_t smem = QDIM * sizeof(float2) + 16 * sizeof(float); // 131136 B
    // Opt in to >64KB dynamic LDS (WGP has 320 KB). Idempotent, capture-safe.
    hipFuncSetAttribute((const void*)qsim14_kernel,
                        hipFuncAttributeMaxDynamicSharedMemorySize, (int)smem);

    qsim14_kernel<<<256, THREADS, smem, stream>>>(state, params, head_w, head_b, out);
}